// GaussianRasterizer_79714593014349
// MI455X (gfx1250) — compile-verified
//
#include <hip/hip_runtime.h>

#define IMG_W 480
#define IMG_H 360
#define NG 256
#define ALPHA_MIN (1.0f/255.0f)
#define ALPHA_MAX 0.99f

typedef __attribute__((ext_vector_type(2))) float v2f;
typedef __attribute__((ext_vector_type(8))) float v8f;

// --------------------------------------------------------------------------
// Kernel 1: stable ascending depth sort of 256 gaussians (O(N^2) rank),
// scatter params SoA into workspace:
//   ws[0..255]=R, [256..]=G, [512..]=B, [768..]=opacity,
//   [1024..]=ca, [1280..]=cb, [1536..]=cc, [1792..]=depth,
//   [2048..]=mx, [2304..]=my   (2560 floats total)
// --------------------------------------------------------------------------
__global__ __launch_bounds__(256) void sort_kernel(const float* __restrict__ md,
                                                   const float* __restrict__ means,
                                                   float* __restrict__ ws) {
    __shared__ float sdep[NG];
    const int tid = threadIdx.x;

    const float r  = md[tid*8+0];
    const float g  = md[tid*8+1];
    const float b  = md[tid*8+2];
    const float op = md[tid*8+3];
    const float ca = md[tid*8+4];
    const float cb = md[tid*8+5];
    const float cc = md[tid*8+6];
    const float dp = md[tid*8+7];
    const float mx = means[tid*2+0];
    const float my = means[tid*2+1];

    sdep[tid] = dp;
    __syncthreads();

    int rank = 0;
#pragma unroll 16
    for (int j = 0; j < NG; ++j) {
        const float dj = sdep[j];
        rank += (dj < dp) || (dj == dp && j < tid);   // stable ascending
    }

    ws[0*NG + rank] = r;
    ws[1*NG + rank] = g;
    ws[2*NG + rank] = b;
    ws[3*NG + rank] = op;
    ws[4*NG + rank] = ca;
    ws[5*NG + rank] = cb;
    ws[6*NG + rank] = cc;
    ws[7*NG + rank] = dp;
    ws[8*NG + rank] = mx;
    ws[9*NG + rank] = my;
}

// --------------------------------------------------------------------------
// Kernel 2: rasterize one 32x8 pixel tile per 256-thread block (8 wave32s,
// each wave owns one 32-pixel row). Power = F(pixel) . G(gaussian) computed
// with v_wmma_f32_16x16x4_f32 (K=6 padded to 8 -> 2 chained WMMAs per tile),
// staged through wave-private LDS (DS ops are in-order per wave), then
// sequential front-to-back compositing per pixel.
// --------------------------------------------------------------------------
__global__ __launch_bounds__(256) void raster_kernel(const float* __restrict__ gsrc,
                                                     float* __restrict__ out) {
    __shared__ float sg[10*NG];      // sorted gaussian params (SoA)
    __shared__ float gfeat[8*NG];    // gaussian features, k-major: gfeat[k*256+n]
    __shared__ float ffeat[256*8];   // pixel features, pixel-major
    __shared__ float ptile[8*512];   // per-wave power staging (2 x 16x16 tiles)

    const int tid  = threadIdx.x;
    const int lane = tid & 31;
    const int wave = tid >> 5;
    const int sel  = lane >> 4;      // which half of the wave (K-pair select)
    const int c    = lane & 15;      // row/col index within 16
    const int k0   = sel * 2;        // VGPR0 holds K=k0, VGPR1 holds K=k0+1

    const int x0 = blockIdx.x * 32;  // tile origin (recentering keeps f32 exact)
    const int y0 = blockIdx.y * 8;
    const int px = x0 + lane;        // this thread's pixel
    const int py = y0 + wave;

    // ---- load sorted params; thread tid owns gaussian tid's fields ----
    float v[10];
#pragma unroll
    for (int i = 0; i < 10; ++i) {
        v[i] = gsrc[i*NG + tid];
        sg[i*NG + tid] = v[i];
    }

    // ---- gaussian features (tile-local coordinates) ----
    {
        const float ca = v[4], cb = v[5], cc = v[6];
        const float lmx = v[8] - (float)x0;
        const float lmy = v[9] - (float)y0;
        gfeat[0*NG + tid] = -0.5f * ca;
        gfeat[1*NG + tid] = -0.5f * cc;
        gfeat[2*NG + tid] = -cb;
        gfeat[3*NG + tid] = ca*lmx + cb*lmy;
        gfeat[4*NG + tid] = cc*lmy + cb*lmx;
        gfeat[5*NG + tid] = -(0.5f*ca*lmx*lmx + cb*lmx*lmy + 0.5f*cc*lmy*lmy);
        gfeat[6*NG + tid] = 0.0f;    // K padding
        gfeat[7*NG + tid] = 0.0f;    // K padding
    }

    // ---- pixel features ----
    {
        const float lx = (float)lane + 0.5f;
        const float ly = (float)wave + 0.5f;
        float* f = &ffeat[tid*8];
        f[0] = lx*lx; f[1] = ly*ly; f[2] = lx*ly;
        f[3] = lx;    f[4] = ly;    f[5] = 1.0f;
        f[6] = 0.0f;  f[7] = 0.0f;
    }
    __syncthreads();

    // ---- build constant A matrices (16 pixels x K) for both pixel halves ----
    const int fb = wave*32*8;
    v2f aLo0 = { ffeat[fb + c*8 + k0    ], ffeat[fb + c*8 + k0 + 1] };
    v2f aHi0 = { ffeat[fb + c*8 + k0 + 4], ffeat[fb + c*8 + k0 + 5] };
    v2f aLo1 = { ffeat[fb + (16+c)*8 + k0    ], ffeat[fb + (16+c)*8 + k0 + 1] };
    v2f aHi1 = { ffeat[fb + (16+c)*8 + k0 + 4], ffeat[fb + (16+c)*8 + k0 + 5] };

    const int pb = wave*512;                                 // wave-private region
    const int rb = pb + (lane < 16 ? 0 : 256) + (lane & 15)*16;  // this lane's pixel row

    float accR = 0.f, accG = 0.f, accB = 0.f, accD = 0.f, T = 1.f;

    for (int chunk = 0; chunk < 16; ++chunk) {
        const int gb = chunk*16 + c;
        v2f bLo = { gfeat[(k0    )*NG + gb], gfeat[(k0 + 1)*NG + gb] };
        v2f bHi = { gfeat[(k0 + 4)*NG + gb], gfeat[(k0 + 5)*NG + gb] };

        // power tiles: D = A_lo x B_lo + A_hi x B_hi  (K = 8, padded from 6)
        v8f d0 = {};
        d0 = __builtin_amdgcn_wmma_f32_16x16x4_f32(false, aLo0, false, bLo,
                                                   (short)0, d0, false, false);
        d0 = __builtin_amdgcn_wmma_f32_16x16x4_f32(false, aHi0, false, bHi,
                                                   (short)0, d0, false, false);
        v8f d1 = {};
        d1 = __builtin_amdgcn_wmma_f32_16x16x4_f32(false, aLo1, false, bLo,
                                                   (short)0, d1, false, false);
        d1 = __builtin_amdgcn_wmma_f32_16x16x4_f32(false, aHi1, false, bHi,
                                                   (short)0, d1, false, false);

        // stage D (pixel-row major) so each lane can read its own pixel's row
#pragma unroll
        for (int r = 0; r < 8; ++r) {
            ptile[pb +       (sel*8 + r)*16 + c] = d0[r];
            ptile[pb + 256 + (sel*8 + r)*16 + c] = d1[r];
        }
        // DS ops are in-order within a wave; just stop the compiler from
        // moving the reads above the writes.
        __builtin_amdgcn_wave_barrier();
        asm volatile("" ::: "memory");

        // read this pixel's 16 powers (contiguous -> 4x ds_load_b128)
        float pv[16];
#pragma unroll
        for (int q = 0; q < 4; ++q) {
            const float4 t = *reinterpret_cast<const float4*>(&ptile[rb + q*4]);
            pv[q*4+0] = t.x; pv[q*4+1] = t.y; pv[q*4+2] = t.z; pv[q*4+3] = t.w;
        }

        // composite 16 gaussians front-to-back (branchless; form is neg-definite)
#pragma unroll
        for (int g = 0; g < 16; ++g) {
            const float pw = pv[g];
            const int   n  = chunk*16 + g;
            float a = fminf(sg[3*NG + n] * __expf(pw), ALPHA_MAX);
            a = (pw > 0.0f)     ? 0.0f : a;
            a = (a < ALPHA_MIN) ? 0.0f : a;
            const float wgt = a * T;
            accR += wgt * sg[0*NG + n];
            accG += wgt * sg[1*NG + n];
            accB += wgt * sg[2*NG + n];
            accD += wgt * sg[7*NG + n];
            T *= (1.0f - a);
        }
        // WAR guard: next chunk's stores must not be hoisted above these reads
        __builtin_amdgcn_wave_barrier();
        asm volatile("" ::: "memory");
    }

    // ---- outputs: color(H,W,3) | depth(H,W) | sil(H,W), concatenated ----
    const int p = py*IMG_W + px;
    out[p*3 + 0] = accR;
    out[p*3 + 1] = accG;
    out[p*3 + 2] = accB;
    out[IMG_H*IMG_W*3 + p] = accD;
    out[IMG_H*IMG_W*4 + p] = 1.0f - T;
}

// --------------------------------------------------------------------------
extern "C" void kernel_launch(void* const* d_in, const int* in_sizes, int n_in,
                              void* d_out, int out_size, void* d_ws, size_t ws_size,
                              hipStream_t stream) {
    (void)in_sizes; (void)n_in; (void)out_size; (void)ws_size;
    const float* md    = (const float*)d_in[0];   // (256, 8) f32
    const float* means = (const float*)d_in[1];   // (256, 2) f32
    float*       ws    = (float*)d_ws;            // 2560 floats used
    float*       out   = (float*)d_out;           // 864000 floats

    sort_kernel<<<1, 256, 0, stream>>>(md, means, ws);
    raster_kernel<<<dim3(IMG_W/32, IMG_H/8), 256, 0, stream>>>(ws, out);
}